// GCN_84937273246040
// MI455X (gfx1250) — compile-verified
//
#include <hip/hip_runtime.h>

// ---------------------------------------------------------------------------
// GCN forward: 3 x ( SpMM(A_norm, H) ; H @ W + b ; ReLU except last )
// N=100000, Fin=128, Fout=128/128/64, nnz ~ 3.3M.
//
// SpMM: edge-parallel, 1 wave per edge, float4 per lane covers the 128-wide
//       feature row; scatter via global_atomic_add_f32. H (51MB) is
//       L2-resident (192MB L2), so gathers don't hit HBM.
// GEMM: V_WMMA_F32_16X16X4_F32 (full fp32 on the matrix pipe).
//       Block = 256 threads (8 waves) covers 64 rows x Fout.
//       Each wave owns one 16-col tile and MT row-tiles; one B fragment is
//       reused across MT WMMAs (4x W-load amortization).
// ---------------------------------------------------------------------------

typedef float v2f __attribute__((ext_vector_type(2)));
typedef float v8f __attribute__((ext_vector_type(8)));

// -------------------------- SpMM (F = 128) ---------------------------------
__global__ void spmm128_kernel(const float* __restrict__ H,
                               const int*   __restrict__ rows,
                               const int*   __restrict__ cols,
                               const float* __restrict__ vals,
                               float*       __restrict__ out,
                               int nnz) {
  int wid  = (int)((blockIdx.x * blockDim.x + threadIdx.x) >> 5);
  int lane = (int)(threadIdx.x & 31u);
  if (wid >= nnz) return;

  int   r = rows[wid];
  int   c = cols[wid];
  float v = vals[wid];

  const float4 h = ((const float4*)(H + (size_t)c * 128))[lane];
  float* dst = out + (size_t)r * 128 + lane * 4;
  atomicAdd(dst + 0, v * h.x);
  atomicAdd(dst + 1, v * h.y);
  atomicAdd(dst + 2, v * h.z);
  atomicAdd(dst + 3, v * h.w);
}

// ------------------- GEMM + bias (+ReLU), Fin = 128 ------------------------
// out[nRows x Fout] = A[nRows x 128] @ W[128 x Fout] + bias ; optional ReLU.
// CT = Fout/16 column tiles. 8 waves: wave -> column tile (wave % CT) and a
// run of MTW = CT/2 row-tiles starting at (wave / CT) * MTW. Block = 64 rows.
#define LDS_STRIDE 132   // 128 + 4 pad: kills 16-way LDS bank conflicts;
                         // 132 floats = 528 B, still 16B-aligned per row.

template <int CT, bool RELU>
__global__ void gemm_bias_kernel(const float* __restrict__ A,
                                 const float* __restrict__ W,
                                 const float* __restrict__ bias,
                                 float*       __restrict__ out,
                                 int nRows) {
  constexpr int Fout = CT * 16;
  constexpr int WPC  = 8 / CT;       // waves per column tile (1 or 2)
  constexpr int MTW  = 4 / WPC;      // row-tiles per wave (4 or 2)

  __shared__ __align__(16) float As[64 * LDS_STRIDE];

  const int row0 = (int)blockIdx.x * 64;
  // Block-uniform (scalar) tail test: nearly every block takes the fast path.
  const bool fullBlock = (row0 + 64 <= nRows);

  // Cooperative float4 load of the 64x128 A tile (256 threads x 8 float4).
  if (fullBlock) {
    for (int i = (int)threadIdx.x; i < 64 * 32; i += (int)blockDim.x) {
      int rr = i >> 5;             // row within tile
      int c4 = (i & 31) << 2;      // k index (multiple of 4)
      *(float4*)&As[rr * LDS_STRIDE + c4] =
          *(const float4*)(A + (size_t)(row0 + rr) * 128 + c4);
    }
  } else {
    for (int i = (int)threadIdx.x; i < 64 * 32; i += (int)blockDim.x) {
      int rr = i >> 5;
      int c4 = (i & 31) << 2;
      int gr = row0 + rr;
      float4 va = make_float4(0.f, 0.f, 0.f, 0.f);
      if (gr < nRows) va = *(const float4*)(A + (size_t)gr * 128 + c4);
      *(float4*)&As[rr * LDS_STRIDE + c4] = va;
    }
  }
  __syncthreads();

  const int wave = (int)(threadIdx.x >> 5);   // 0..7
  const int lane = (int)(threadIdx.x & 31u);
  const int l16  = lane & 15;
  const int hi   = lane >> 4;                 // 0: lanes 0-15, 1: lanes 16-31
  const int koff = hi << 1;                   // K sub-offset 0 or 2

  const int ct   = wave % CT;                 // column tile
  const int col0 = ct * 16;
  const int mt0  = (wave / CT) * MTW;         // first row-tile for this wave

  v8f c[MTW];
#pragma unroll
  for (int mt = 0; mt < MTW; ++mt) c[mt] = (v8f){};

#pragma unroll 4
  for (int k = 0; k < 128; k += 4) {
    // B fragment (4x16 f32): N = lane%16, VGPR0/1 = K = k+koff / k+koff+1
    v2f b;
    b.x = W[(size_t)(k + koff)     * Fout + col0 + l16];
    b.y = W[(size_t)(k + koff + 1) * Fout + col0 + l16];
#pragma unroll
    for (int mt = 0; mt < MTW; ++mt) {
      // A fragment (16x4 f32): M = lane%16 within row-tile mt0+mt
      const int ar = (mt0 + mt) * 16 + l16;
      v2f a;
      a.x = As[ar * LDS_STRIDE + k + koff];
      a.y = As[ar * LDS_STRIDE + k + koff + 1];
      // (neg_a, A, neg_b, B, c_mod, C, reuse_a, reuse_b)
      c[mt] = __builtin_amdgcn_wmma_f32_16x16x4_f32(false, a, false, b,
                                                    (short)0, c[mt], false, false);
    }
  }

  // C/D layout: VGPR r -> M = r + 8*hi, N = lane%16
  const int   n     = l16;
  const int   mbase = hi << 3;
  const float bv    = bias[col0 + n];

  if (fullBlock) {
    // Scalar-branch fast path: straight-line unguarded stores.
#pragma unroll
    for (int mt = 0; mt < MTW; ++mt) {
      const int tileBase = row0 + (mt0 + mt) * 16;
#pragma unroll
      for (int r = 0; r < 8; ++r) {
        float o = c[mt][r] + bv;
        if (RELU) o = fmaxf(o, 0.0f);
        out[(size_t)(tileBase + mbase + r) * Fout + col0 + n] = o;
      }
    }
  } else {
    // Tail block (at most one per launch): guarded stores.
#pragma unroll
    for (int mt = 0; mt < MTW; ++mt) {
      const int tileBase = row0 + (mt0 + mt) * 16;
#pragma unroll
      for (int r = 0; r < 8; ++r) {
        int m = tileBase + mbase + r;
        if (m < nRows) {
          float o = c[mt][r] + bv;
          if (RELU) o = fmaxf(o, 0.0f);
          out[(size_t)m * Fout + col0 + n] = o;
        }
      }
    }
  }
}

// ---------------------------------------------------------------------------
extern "C" void kernel_launch(void* const* d_in, const int* in_sizes, int n_in,
                              void* d_out, int out_size, void* d_ws, size_t ws_size,
                              hipStream_t stream) {
  const float* X    = (const float*)d_in[0];
  const int*   rows = (const int*)  d_in[1];
  const int*   cols = (const int*)  d_in[2];
  const float* vals = (const float*)d_in[3];
  const float* W1   = (const float*)d_in[4];
  const float* b1   = (const float*)d_in[5];
  const float* W2   = (const float*)d_in[6];
  const float* b2   = (const float*)d_in[7];
  const float* W3   = (const float*)d_in[8];
  const float* b3   = (const float*)d_in[9];
  float*       out  = (float*)d_out;

  const int N    = in_sizes[0] / 128;   // 100000
  const int nnz  = in_sizes[1];         // ~3.3M
  const int OUTD = in_sizes[8] / 128;   // 64

  float* T = (float*)d_ws;                 // SpMM accumulator, N x 128
  float* H = T + (size_t)N * 128;          // hidden activations, N x 128
  const size_t bufBytes = (size_t)N * 128 * sizeof(float);

  const dim3 blk(256);
  const int spmmBlocks = (nnz + 7) / 8;    // 8 waves / block, 1 edge / wave
  const int rowBlocks  = (N + 63) / 64;    // 64 rows per GEMM block

  // Layer 1: T = A_norm @ X ; H = relu(T @ W1 + b1)
  hipMemsetAsync(T, 0, bufBytes, stream);
  spmm128_kernel<<<spmmBlocks, blk, 0, stream>>>(X, rows, cols, vals, T, nnz);
  gemm_bias_kernel<8, true><<<rowBlocks, blk, 0, stream>>>(T, W1, b1, H, N);

  // Layer 2: T = A_norm @ H ; H = relu(T @ W2 + b2)
  hipMemsetAsync(T, 0, bufBytes, stream);
  spmm128_kernel<<<spmmBlocks, blk, 0, stream>>>(H, rows, cols, vals, T, nnz);
  gemm_bias_kernel<8, true><<<rowBlocks, blk, 0, stream>>>(T, W2, b2, H, N);

  // Layer 3: T = A_norm @ H ; out = T @ W3 + b3   (no ReLU)
  hipMemsetAsync(T, 0, bufBytes, stream);
  spmm128_kernel<<<spmmBlocks, blk, 0, stream>>>(H, rows, cols, vals, T, nnz);
  if (OUTD == 64) {
    gemm_bias_kernel<4, false><<<rowBlocks, blk, 0, stream>>>(T, W3, b3, out, N);
  } else {
    gemm_bias_kernel<8, false><<<rowBlocks, blk, 0, stream>>>(T, W3, b3, out, N);
  }
}